// Lapla_filter_77584289235642
// MI455X (gfx1250) — compile-verified
//
#include <hip/hip_runtime.h>

// ---------------------------------------------------------------------------
// GCN Laplacian filter for MI455X (gfx1250, wave32).
//   feat = relu(x@W1+b1)                    -> WMMA f32 16x16x4 GEMM
//   agg[dst] += feat[src]*dinv[src]         -> wave-per-edge, f32 L2 atomics
//   out  = (feat - agg*dinv)@W2 + b2        -> WMMA f32 16x16x4 GEMM
// feat is staged in d_out (overwritten in-place by GEMM2, block-row safe).
// ---------------------------------------------------------------------------

typedef float v2f __attribute__((ext_vector_type(2)));
typedef float v8f __attribute__((ext_vector_type(8)));

#define DF 128           // feature width (D_IN == D_HID == D_OUT == 128)
#define LDSW 132         // padded LDS row stride (conflict-free A reads)

// ------------------------- utility kernels ---------------------------------

__global__ __launch_bounds__(256) void zero_kernel(float* p, long long n) {
  long long i = (long long)blockIdx.x * blockDim.x + threadIdx.x;
  long long stride = (long long)gridDim.x * blockDim.x;
  for (; i < n; i += stride) p[i] = 0.0f;
}

__global__ __launch_bounds__(256) void degree_kernel(const int* __restrict__ dst,
                                                     float* __restrict__ deg,
                                                     int n_edges) {
  int i = blockIdx.x * blockDim.x + threadIdx.x;
  int stride = gridDim.x * blockDim.x;
  for (; i < n_edges; i += stride) atomicAdd(&deg[dst[i]], 1.0f);
}

__global__ __launch_bounds__(256) void dinv_kernel(const float* __restrict__ deg,
                                                   float* __restrict__ dinv, int n) {
  int i = blockIdx.x * blockDim.x + threadIdx.x;
  if (i < n) dinv[i] = rsqrtf(fmaxf(deg[i], 1.0f));
}

// ------------------- edge scatter: one wave32 per edge ----------------------
// lanes cover 128 channels as float4; src/dst are wave-uniform -> scalarizable.

__global__ __launch_bounds__(256) void edge_kernel(const float4* __restrict__ feat4,
                                                   const int* __restrict__ src,
                                                   const int* __restrict__ dst,
                                                   const float* __restrict__ dinv,
                                                   float* __restrict__ agg,
                                                   int n_edges) {
  const int lane = threadIdx.x & 31;
  int wave = blockIdx.x * (blockDim.x >> 5) + (threadIdx.x >> 5);
  const int nwaves = gridDim.x * (blockDim.x >> 5);
  for (int e = wave; e < n_edges; e += nwaves) {
    const int s = src[e];
    const int d = dst[e];
    const float w = dinv[s];
    const float4 f = feat4[(long long)s * (DF / 4) + lane];
    float* base = agg + (long long)d * DF + lane * 4;
    atomicAdd(base + 0, f.x * w);   // global_atomic_add_f32, no return
    atomicAdd(base + 1, f.y * w);
    atomicAdd(base + 2, f.z * w);
    atomicAdd(base + 3, f.w * w);
  }
}

// --------------------------- WMMA GEMM cores --------------------------------
// Per block: 16 output rows. 8 waves, wave w owns columns [16w, 16w+16).
// A fragment (16x4 f32): lanes 0-15 -> K=k,k+1 ; lanes 16-31 -> K=k+2,k+3.
// B fragment (4x16 f32): mirrored. C/D: 8 VGPRs, M = r + 8*(lane>=16).

__device__ __forceinline__ v8f wmma_tile_f32(const float* As, const float* __restrict__ W,
                                             int lane, int n0) {
  const int col  = lane & 15;
  const int koff = (lane >> 4) * 2;   // 0 or 2
  v8f acc = {0.f, 0.f, 0.f, 0.f, 0.f, 0.f, 0.f, 0.f};
#pragma unroll
  for (int k = 0; k < DF; k += 4) {
    v2f a, b;
    a.x = As[col * LDSW + k + koff];
    a.y = As[col * LDSW + k + koff + 1];
    b.x = W[(k + koff) * DF + n0 + col];
    b.y = W[(k + koff + 1) * DF + n0 + col];
    acc = __builtin_amdgcn_wmma_f32_16x16x4_f32(false, a, false, b,
                                                (short)0, acc, false, false);
  }
  return acc;
}

// feat = relu(x @ W1 + b1)
__global__ __launch_bounds__(256) void gemm_relu_kernel(const float* __restrict__ X,
                                                        const float* __restrict__ W,
                                                        const float* __restrict__ bias,
                                                        float* __restrict__ out,
                                                        int n_rows) {
  __shared__ float As[16 * LDSW];
  const int m0 = blockIdx.x * 16;
  const int tid = threadIdx.x;
  for (int i = tid; i < 16 * DF; i += 256) {
    const int r = i >> 7, c = i & (DF - 1);
    const int row = m0 + r;
    As[r * LDSW + c] = (row < n_rows) ? X[(long long)row * DF + c] : 0.0f;
  }
  __syncthreads();

  const int lane = tid & 31;
  const int n0 = (tid >> 5) * 16;
  v8f acc = wmma_tile_f32(As, W, lane, n0);

  const int col = n0 + (lane & 15);
  const float bb = bias[col];
#pragma unroll
  for (int r = 0; r < 8; ++r) {
    const int row = m0 + r + ((lane >> 4) * 8);
    if (row < n_rows)
      out[(long long)row * DF + col] = fmaxf(acc[r] + bb, 0.0f);
  }
}

// out = (feat - agg*dinv) @ W2 + b2   (feat aliases out: read tile, sync, write)
__global__ __launch_bounds__(256) void gemm_final_kernel(const float* FEAT,
                                                         const float* __restrict__ AGG,
                                                         const float* __restrict__ DINV,
                                                         const float* __restrict__ W,
                                                         const float* __restrict__ bias,
                                                         float* out,
                                                         int n_rows) {
  __shared__ float As[16 * LDSW];
  const int m0 = blockIdx.x * 16;
  const int tid = threadIdx.x;
  for (int i = tid; i < 16 * DF; i += 256) {
    const int r = i >> 7, c = i & (DF - 1);
    const int row = m0 + r;
    float v = 0.0f;
    if (row < n_rows) {
      const long long idx = (long long)row * DF + c;
      v = FEAT[idx] - AGG[idx] * DINV[row];
    }
    As[r * LDSW + c] = v;
  }
  __syncthreads();

  const int lane = tid & 31;
  const int n0 = (tid >> 5) * 16;
  v8f acc = wmma_tile_f32(As, W, lane, n0);

  const int col = n0 + (lane & 15);
  const float bb = bias[col];
#pragma unroll
  for (int r = 0; r < 8; ++r) {
    const int row = m0 + r + ((lane >> 4) * 8);
    if (row < n_rows)
      out[(long long)row * DF + col] = acc[r] + bb;
  }
}

// ------------------------------- launcher -----------------------------------

extern "C" void kernel_launch(void* const* d_in, const int* in_sizes, int n_in,
                              void* d_out, int out_size, void* d_ws, size_t ws_size,
                              hipStream_t stream) {
  const float* x   = (const float*)d_in[0];
  const int*   src = (const int*)d_in[1];
  const int*   dst = (const int*)d_in[2];
  const float* W1  = (const float*)d_in[3];
  const float* b1  = (const float*)d_in[4];
  const float* W2  = (const float*)d_in[5];
  const float* b2  = (const float*)d_in[6];

  const int n_nodes = in_sizes[0] / DF;
  const int n_edges = in_sizes[1];

  float* out  = (float*)d_out;                    // also holds feat temporarily
  float* agg  = (float*)d_ws;                     // n_nodes * 128
  float* deg  = agg + (long long)n_nodes * DF;    // n_nodes
  float* dinv = deg + n_nodes;                    // n_nodes

  const long long zero_n = (long long)n_nodes * DF + n_nodes;  // agg + deg
  const int mblocks = (n_nodes + 15) / 16;

  zero_kernel      <<<4096, 256, 0, stream>>>(agg, zero_n);
  degree_kernel    <<<2048, 256, 0, stream>>>(dst, deg, n_edges);
  dinv_kernel      <<<(n_nodes + 255) / 256, 256, 0, stream>>>(deg, dinv, n_nodes);
  gemm_relu_kernel <<<mblocks, 256, 0, stream>>>(x, W1, b1, out, n_nodes);
  edge_kernel      <<<8192, 256, 0, stream>>>((const float4*)out, src, dst, dinv,
                                              agg, n_edges);
  gemm_final_kernel<<<mblocks, 256, 0, stream>>>(out, agg, dinv, W2, b2, out,
                                                 n_nodes);
}